// LSTM_75565654605935
// MI455X (gfx1250) — compile-verified
//
#include <hip/hip_runtime.h>

typedef float v2f __attribute__((ext_vector_type(2)));
typedef float v8f __attribute__((ext_vector_type(8)));

#define SEQ 128
#define IN  512
#define HID 12

// Hardware V_TANH_F32 (gfx1250 TRANS op) when the builtin exists; libm otherwise.
__device__ __forceinline__ float fast_tanh(float a) {
#if __has_builtin(__builtin_amdgcn_tanhf)
    return __builtin_amdgcn_tanhf(a);
#else
    return tanhf(a);
#endif
}

// ---------------------------------------------------------------------------
// Prep: transpose + pad input-projection weights into ws:
//   Wt [i*16 + h] = (h<12) ? w_ih_f[h*512 + i] : 0     (512x16 f32, 32 KB)
//   Wrt[i*16 + h] = (h<12) ? w_ih_r[h*512 + i] : 0     (512x16 f32, 32 KB)
// Makes every WMMA B-fragment load a contiguous 64B row.
// ---------------------------------------------------------------------------
__global__ __launch_bounds__(256) void prep_kernel(const float* __restrict__ wf,
                                                   const float* __restrict__ wr,
                                                   float* __restrict__ ws) {
    int t = blockIdx.x * blockDim.x + threadIdx.x;   // 0 .. 512*16-1
    if (t >= IN * 16) return;
    int i = t >> 4;
    int h = t & 15;
    float vf = (h < HID) ? wf[h * IN + i] : 0.0f;
    float vr = (h < HID) ? wr[h * IN + i] : 0.0f;
    ws[t]           = vf;
    ws[IN * 16 + t] = vr;
}

// ---------------------------------------------------------------------------
// Fused kernel: one wave32 per batch element.
//  Phase 1: gathered GEMM pre[s,h] = sum_i emb[x[b,i],s] * W[h,i]
//           via V_WMMA_F32_16X16X4_F32, M=s(128 -> 8 tiles), N=h(16), K=i(512).
//           acc[8] (v8f each) holds the full 128x16 "pre" in registers.
//           acc2 accumulates the reverse weights; only row m=127 is used.
//  Phase 2: 128-step serial tanh recurrence, lanes = hidden units, h-vector
//           broadcast through LDS each step (single-wave workgroup).
//  Phase 3: h_bwd tanh + 24->3 FC epilogue.
// ---------------------------------------------------------------------------
__global__ __launch_bounds__(32) void rnn_kernel(
    const int*   __restrict__ x,     const float* __restrict__ emb,
    const float* __restrict__ Wt,    const float* __restrict__ Wrt,
    const float* __restrict__ whh,   const float* __restrict__ bihf,
    const float* __restrict__ bhhf,  const float* __restrict__ bihr,
    const float* __restrict__ bhhr,  const float* __restrict__ fcw,
    const float* __restrict__ fcb,   float* __restrict__ out) {

    const int  b    = blockIdx.x;          // uniform -> scalar loads for indices
    const int  lane = threadIdx.x;         // 0..31 (wave32)
    const int  lh   = lane & 15;
    const bool hiH  = lane >= 16;

    __shared__ float pre[SEQ * 16];        // pre[s*16 + h], 8 KB
    __shared__ float hS[16];               // shared hidden vector per step
    __shared__ float hbS[16];              // reverse preactivation (row s=127)

    // ---- Phase 1: WMMA GEMM ------------------------------------------------
    v8f zero = {0.f, 0.f, 0.f, 0.f, 0.f, 0.f, 0.f, 0.f};
    v8f acc[8];
#pragma unroll
    for (int m = 0; m < 8; ++m) acc[m] = zero;
    v8f acc2 = zero;

    const int4* xb = (const int4*)(x + b * IN);

    for (int kt = 0; kt < IN / 4; ++kt) {
        int4 idx = xb[kt];                 // scalar s_load_b128 (b uniform)

        // f32 16x16x4 A layout: lanes 0-15 -> K=0 (v0), K=1 (v1);
        //                       lanes 16-31 -> K=2 (v0), K=3 (v1)
        int rA = hiH ? idx.z : idx.x;
        int rB = hiH ? idx.w : idx.y;
        const float* pa = emb + rA * SEQ + lh;   // 64B coalesced per half-wave
        const float* pb = emb + rB * SEQ + lh;

        // B layout mirrors A: v0 = K{0|2}, v1 = K{1|3}; lane = N = h
        const float* w0 = Wt  + kt * 64 + (hiH ? 32 : 0) + lh;
        const float* w1 = Wrt + kt * 64 + (hiH ? 32 : 0) + lh;
        v2f bfrag, brfrag;
        bfrag.x  = w0[0];   bfrag.y  = w0[16];
        brfrag.x = w1[0];   brfrag.y = w1[16];

#pragma unroll
        for (int mt = 0; mt < 8; ++mt) {
            v2f afrag;
            afrag.x = pa[mt * 16];
            afrag.y = pb[mt * 16];
            acc[mt] = __builtin_amdgcn_wmma_f32_16x16x4_f32(
                false, afrag, false, bfrag, (short)0, acc[mt], false, false);
            if (mt == 7)  // reverse direction shares the last A tile (s=112..127)
                acc2 = __builtin_amdgcn_wmma_f32_16x16x4_f32(
                    false, afrag, false, brfrag, (short)0, acc2, false, false);
        }
    }

    // Dump pre to LDS. C/D layout: VGPR r: lanes0-15 -> M=r, lanes16-31 -> M=r+8
#pragma unroll
    for (int mt = 0; mt < 8; ++mt) {
#pragma unroll
        for (int r = 0; r < 8; ++r) {
            int s = mt * 16 + r + (hiH ? 8 : 0);
            pre[s * 16 + lh] = acc[mt][r];
        }
    }
    if (hiH) hbS[lh] = acc2[7];            // m = 112 + 15 = 127, h = lane-16

    // ---- Phase 2: serial tanh recurrence (lane = hidden unit) --------------
    const int myi = (lh < HID) ? lh : 0;   // lanes 12-15 (and clones) are benign
    float wrow[HID];
#pragma unroll
    for (int j = 0; j < HID; ++j) wrow[j] = whh[myi * HID + j];
    const float bias = bihf[myi] + bhhf[myi];

    float h[HID];
#pragma unroll
    for (int j = 0; j < HID; ++j) h[j] = 0.0f;

    __syncthreads();                        // pre/hbS visible

    for (int s = 0; s < SEQ; ++s) {
        // 4 parallel partial sums -> dependent chain of 3 FMAs + 2 adds
        float p  = pre[s * 16 + myi] + bias;
        float t0 = fmaf(wrow[0], h[0], fmaf(wrow[4], h[4], fmaf(wrow[8],  h[8],  p)));
        float t1 = fmaf(wrow[1], h[1], fmaf(wrow[5], h[5], wrow[9]  * h[9]));
        float t2 = fmaf(wrow[2], h[2], fmaf(wrow[6], h[6], wrow[10] * h[10]));
        float t3 = fmaf(wrow[3], h[3], fmaf(wrow[7], h[7], wrow[11] * h[11]));
        float nh = fast_tanh((t0 + t1) + (t2 + t3));

        __syncthreads();                    // previous hS fully consumed (S_NOP in HW)
        hS[lh] = nh;                        // lanes L and L+16 write identical values
        __syncthreads();
#pragma unroll
        for (int j = 0; j < HID; ++j) h[j] = hS[j];
    }

    // ---- Phase 3: reverse hidden + FC --------------------------------------
    float hb[HID];
#pragma unroll
    for (int j = 0; j < HID; ++j) hb[j] = fast_tanh(hbS[j] + bihr[j] + bhhr[j]);

    float o[3];
#pragma unroll
    for (int oo = 0; oo < 3; ++oo) {
        float a = fcb[oo];
#pragma unroll
        for (int j = 0; j < HID; ++j) {
            a = fmaf(fcw[oo * 24 + j],        h[j],  a);
            a = fmaf(fcw[oo * 24 + HID + j],  hb[j], a);
        }
        o[oo] = a;
    }
    if (lane < 3)
        out[b * 3 + lane] = (lane == 0) ? o[0] : ((lane == 1) ? o[1] : o[2]);
}

// ---------------------------------------------------------------------------
extern "C" void kernel_launch(void* const* d_in, const int* in_sizes, int n_in,
                              void* d_out, int out_size, void* d_ws, size_t ws_size,
                              hipStream_t stream) {
    (void)in_sizes; (void)n_in; (void)out_size; (void)ws_size;

    const int*   x      = (const int*)  d_in[0];
    const float* emb    = (const float*)d_in[1];
    const float* w_ih_f = (const float*)d_in[2];
    const float* w_hh_f = (const float*)d_in[3];
    const float* b_ih_f = (const float*)d_in[4];
    const float* b_hh_f = (const float*)d_in[5];
    const float* w_ih_r = (const float*)d_in[6];
    /* d_in[7] = w_hh_r is dead in the reference (single reverse step from h=0) */
    const float* b_ih_r = (const float*)d_in[8];
    const float* b_hh_r = (const float*)d_in[9];
    const float* fc_w   = (const float*)d_in[10];
    const float* fc_b   = (const float*)d_in[11];
    float*       outp   = (float*)d_out;

    float* Wt  = (float*)d_ws;        // 512*16 floats
    float* Wrt = Wt + IN * 16;        // 512*16 floats  (total 64 KB of ws)

    prep_kernel<<<(IN * 16 + 255) / 256, 256, 0, stream>>>(w_ih_f, w_ih_r, Wt);
    rnn_kernel<<<2048, 32, 0, stream>>>(x, emb, Wt, Wrt, w_hh_f, b_ih_f, b_hh_f,
                                        b_ih_r, b_hh_r, fc_w, fc_b, outp);
}